// MWEWordLevelSkipGramTaskModel_75668733821512
// MI455X (gfx1250) — compile-verified
//
#include <hip/hip_runtime.h>
#include <hip/hip_bf16.h>

// MI455X / gfx1250, wave32. Gather-bound SGNS loss; f32 WMMA 16x16x4 chains
// compute per-tile Gram diagonals at full precision. Index loads are
// software-pipelined one iteration ahead so the dependent index->row-load
// round trip hides under the previous WMMA chain.

#define VOCAB 500000
#define EMB_D 128
#define NB    16384
#define NK    10
#define NB2   8192
#define NL    5
#define NW    10

typedef __attribute__((ext_vector_type(2))) float v2f;
typedef __attribute__((ext_vector_type(8))) float v8f;

// softplus(x) = log(1 + exp(x)), numerically stable; -log_sigmoid(x) = softplus(-x)
__device__ __forceinline__ float softplusf(float x) {
    return fmaxf(x, 0.f) + __logf(1.f + __expf(-fabsf(x)));
}

__device__ __forceinline__ float wave_reduce_add(float v) {
#pragma unroll
    for (int off = 16; off > 0; off >>= 1)
        v += __shfl_xor(v, off, 32);
    return v;
}

// One 128-deep diagonal-dot chain: acc += A_tile(16x128) x B_tile(128x16)
// restricted to the K-chunk layout (lane l: row l%16, float2 at 4c + 2*(l/16)).
__device__ __forceinline__ v8f dot_chain(const v2f* __restrict__ a,
                                         const float* __restrict__ brow,
                                         int half) {
    v8f acc = {};
#pragma unroll
    for (int c = 0; c < 32; ++c) {
        v2f bb = *reinterpret_cast<const v2f*>(brow + 4 * c + 2 * half);
        acc = __builtin_amdgcn_wmma_f32_16x16x4_f32(
            false, a[c], false, bb, (short)0, acc, false, false);
    }
    return acc;
}

// ---------------------------------------------------------------------------
// Word-level SGNS: one wave handles a tile of 16 batch rows.
// ---------------------------------------------------------------------------
__global__ __launch_bounds__(256) void sgns_word_kernel(
    const float* __restrict__ center_table,
    const float* __restrict__ context_table,
    const int*   __restrict__ w_center,
    const int*   __restrict__ w_outside,
    const int*   __restrict__ w_neg,
    float*       __restrict__ acc_loss)
{
    const int lane = threadIdx.x & 31;
    const int wave = blockIdx.x * (blockDim.x >> 5) + (threadIdx.x >> 5);
    const int b0   = wave * 16;
    const int row  = lane & 15;   // matrix row / column this lane services
    const int half = lane >> 4;   // K sub-half per ISA f32 A/B layout
    const int b    = b0 + row;

    // Load A fragments: lane holds float2 of ce row (l%16) at col 4c + 2*half.
    const float* arow = center_table + (size_t)w_center[b] * EMB_D;
    v2f a[32];
#pragma unroll
    for (int c = 0; c < 32; ++c)
        a[c] = *reinterpret_cast<const v2f*>(arow + 4 * c + 2 * half);

    // Diagonal ownership: element (e,e) -> VGPR e%8, lane e + 16*(e/8)
    const bool owner = (lane < 8) || (lane >= 24);
    const int  eidx  = (lane < 8) ? lane : ((lane >= 24) ? (lane - 24) : 0);

    float lsum = 0.f;
    int idx = w_outside[b];           // index for j=0
#pragma unroll 1
    for (int j = 0; j < NK + 1; ++j) {
        const float* brow = context_table + (size_t)idx * EMB_D;
        // Issue next index load now; its wait hides under the WMMA chain.
        const int idx_next = (j < NK) ? w_neg[(size_t)b * NK + j] : 0;

        const v8f acc = dot_chain(a, brow, half);

        const float s    = owner ? acc[eidx] : 0.f;
        const float term = (j == 0) ? softplusf(-s) : softplusf(s);
        lsum += owner ? term : 0.f;
        idx = idx_next;
    }

    lsum = wave_reduce_add(lsum);
    if (lane == 0) atomicAdd(acc_loss, lsum);
}

// ---------------------------------------------------------------------------
// MWE SGNS: one wave handles 16 entities. Masked-mean built in-register as the
// A tile, then 10 (window) x 11 (pos+neg) diagonal WMMA chains reuse it.
// ---------------------------------------------------------------------------
__global__ __launch_bounds__(256) void sgns_mwe_kernel(
    const float* __restrict__ center_table,
    const float* __restrict__ context_table,
    const int*   __restrict__ m_center,
    const int*   __restrict__ m_len,
    const int*   __restrict__ m_outside,
    const int*   __restrict__ m_neg,
    float*       __restrict__ acc_loss,
    float*       __restrict__ acc_cnt)
{
    const int lane = threadIdx.x & 31;
    const int wave = blockIdx.x * (blockDim.x >> 5) + (threadIdx.x >> 5);
    const int g0   = wave * 16;
    const int row  = lane & 15;
    const int half = lane >> 4;
    const int b2   = g0 + row;

    // Masked mean over the first m_len tokens -> A fragments.
    const int len = m_len[b2];
    v2f a[32];
#pragma unroll
    for (int c = 0; c < 32; ++c) { a[c].x = 0.f; a[c].y = 0.f; }
#pragma unroll
    for (int l = 0; l < NL; ++l) {
        if (l < len) {
            const float* r =
                center_table + (size_t)m_center[(size_t)b2 * NL + l] * EMB_D;
#pragma unroll
            for (int c = 0; c < 32; ++c)
                a[c] += *reinterpret_cast<const v2f*>(r + 4 * c + 2 * half);
        }
    }
    const float invlen = 1.f / (float)len;
#pragma unroll
    for (int c = 0; c < 32; ++c) a[c] *= invlen;

    const bool owner = (lane < 8) || (lane >= 24);
    const int  eidx  = (lane < 8) ? lane : ((lane >= 24) ? (lane - 24) : 0);

    float lsum = 0.f;
    float vcnt = 0.f;
    int oidx = m_outside[(size_t)b2 * NW];   // window position 0
#pragma unroll 1
    for (int w = 0; w < NW; ++w) {
        const float  vf = (oidx != 0) ? 1.f : 0.f;   // pad token == 0
        const size_t n  = (size_t)b2 * NW + w;
        // Pipeline next window's outside index across this window's work.
        const int oidx_next = (w < NW - 1) ? m_outside[(size_t)b2 * NW + w + 1] : 0;

        int idx = oidx;                               // j=0 uses outside row
#pragma unroll 1
        for (int j = 0; j < NK + 1; ++j) {
            const float* brow = context_table + (size_t)idx * EMB_D;
            const int idx_next = (j < NK) ? m_neg[n * NK + j] : 0;

            const v8f acc = dot_chain(a, brow, half);

            const float s    = owner ? acc[eidx] : 0.f;
            const float term = (j == 0) ? softplusf(-s) : softplusf(s);
            lsum += owner ? (vf * term) : 0.f;
            idx = idx_next;
        }
        vcnt += owner ? vf : 0.f;
        oidx = oidx_next;
    }

    lsum = wave_reduce_add(lsum);
    vcnt = wave_reduce_add(vcnt);
    if (lane == 0) {
        atomicAdd(acc_loss, lsum);
        atomicAdd(acc_cnt, vcnt);
    }
}

__global__ void zero3_kernel(float* __restrict__ ws) {
    if (threadIdx.x < 3) ws[threadIdx.x] = 0.f;
}

__global__ void finalize_kernel(const float* __restrict__ ws,
                                float* __restrict__ out) {
    if (threadIdx.x == 0) {
        const float loss_words = ws[0] / (float)NB;
        const float loss_mwes  = ws[1] / fmaxf(ws[2], 1.f);
        out[0] = loss_words + 25.f * loss_mwes;
    }
}

extern "C" void kernel_launch(void* const* d_in, const int* in_sizes, int n_in,
                              void* d_out, int out_size, void* d_ws, size_t ws_size,
                              hipStream_t stream) {
    const float* center_table  = (const float*)d_in[0];
    const float* context_table = (const float*)d_in[1];
    const int*   w_center      = (const int*)d_in[2];
    const int*   w_outside     = (const int*)d_in[3];
    const int*   w_neg         = (const int*)d_in[4];
    const int*   m_center      = (const int*)d_in[5];
    const int*   m_len         = (const int*)d_in[6];
    const int*   m_outside     = (const int*)d_in[7];
    const int*   m_neg         = (const int*)d_in[8];

    float* ws  = (float*)d_ws;    // [0]=word loss sum, [1]=mwe loss sum, [2]=valid count
    float* out = (float*)d_out;

    zero3_kernel<<<1, 32, 0, stream>>>(ws);

    // Word level: NB/16 = 1024 tiles, 8 waves (256 thr) per block -> 128 blocks
    sgns_word_kernel<<<NB / 16 / 8, 256, 0, stream>>>(
        center_table, context_table, w_center, w_outside, w_neg, ws + 0);

    // MWE level: NB2/16 = 512 tiles -> 64 blocks
    sgns_mwe_kernel<<<NB2 / 16 / 8, 256, 0, stream>>>(
        center_table, context_table, m_center, m_len, m_outside, m_neg,
        ws + 1, ws + 2);

    finalize_kernel<<<1, 32, 0, stream>>>(ws, out);
}